// VectorQuantizer_82197084111269
// MI455X (gfx1250) — compile-verified
//
#include <hip/hip_runtime.h>

typedef __attribute__((ext_vector_type(2))) float v2f;
typedef __attribute__((ext_vector_type(8))) float v8f;

#define K_CODES 8192
#define D_DIM   64
#define HW      1024      // 32*32
#define CH_IMG  65536     // 64*1024 floats per image
#define OUT_ELEMS 1048576 // 16*64*32*32

// ---- pass 1: codebook row norms -> d_ws ----
__global__ __launch_bounds__(256) void vq_norms_kernel(const float* __restrict__ w,
                                                       float* __restrict__ norms) {
  int k = blockIdx.x * blockDim.x + threadIdx.x;
  if (k >= K_CODES) return;
  const float4* r = (const float4*)(w + k * D_DIM);
  float s = 0.f;
#pragma unroll
  for (int i = 0; i < 16; ++i) {
    float4 v = r[i];
    s += v.x * v.x + v.y * v.y + v.z * v.z + v.w * v.w;
  }
  norms[k] = s;
}

// ---- pass 0: zero the two loss accumulators (d_out is poisoned) ----
__global__ void vq_zero_losses(float* __restrict__ out) {
  if (threadIdx.x < 2) out[OUT_ELEMS + threadIdx.x] = 0.f;
}

// ---- main: fp32 WMMA distance matmul + argmin + gather + losses ----
// One wave32 handles 16 z-rows against all 8192 codes.
// WMMA operands: A = 16 codes x 4 dims (f32), B = 4 dims x 16 z-rows (f32).
__global__ __launch_bounds__(256) void vq_main_kernel(const float* __restrict__ z,
                                                      const float* __restrict__ w,
                                                      const float* __restrict__ norms,
                                                      float* __restrict__ out) {
  const int lane = threadIdx.x & 31;
  const int wid  = (blockIdx.x * blockDim.x + threadIdx.x) >> 5;
  const int half = lane >> 4;           // 0: dims {4c,4c+1}; 1: dims {4c+2,4c+3}
  const int row  = wid * 16 + (lane & 15);
  const int b    = row >> 10;           // 1024 rows per image, 16 | 1024
  const int hw   = row & 1023;
  const float* zbase = z + b * CH_IMG + hw;

  // Preload this wave's zf tile as B fragments (dims strided by HW in memory).
  float zb_x[16], zb_y[16];
#pragma unroll
  for (int c = 0; c < 16; ++c) {
    int d0 = 4 * c + 2 * half;
    zb_x[c] = zbase[d0 * HW];
    zb_y[c] = zbase[(d0 + 1) * HW];
  }

  float bestVal = 3.4e38f;
  int   bestIdx = 0;

  for (int kt = 0; kt < K_CODES / 16; ++kt) {
    const float2* wrow = (const float2*)(w + (kt * 16 + (lane & 15)) * D_DIM);

    // prefetch next tile's row for this lane (speculative; dropped if OOB)
    int ktn = (kt + 1 < K_CODES / 16) ? kt + 1 : kt;
    __builtin_prefetch((const void*)(w + (ktn * 16 + (lane & 15)) * D_DIM), 0, 3);

    v8f acc = {0.f, 0.f, 0.f, 0.f, 0.f, 0.f, 0.f, 0.f};
#pragma unroll
    for (int c = 0; c < 16; ++c) {
      float2 av = wrow[2 * c + half];   // dims {4c+2h, 4c+2h+1} of code (lane&15)
      v2f afrag; afrag.x = av.x;   afrag.y = av.y;
      v2f bfrag; bfrag.x = zb_x[c]; bfrag.y = zb_y[c];
      acc = __builtin_amdgcn_wmma_f32_16x16x4_f32(false, afrag, false, bfrag,
                                                  (short)0, acc, false, false);
    }

    // acc[r]: S[code = kt*16 + r + 8*half][zrow = lane&15]
    const int kbase = kt * 16 + 8 * half;
    const float4* np4 = (const float4*)(norms + kbase);
    float4 n0 = np4[0], n1 = np4[1];
    float nv[8] = {n0.x, n0.y, n0.z, n0.w, n1.x, n1.y, n1.z, n1.w};
#pragma unroll
    for (int r = 0; r < 8; ++r) {
      float cand = nv[r] - 2.0f * acc[r];   // ||w||^2 - 2 z.w  (||z||^2 const per row)
      int   idx  = kbase + r;
      bool better = (cand < bestVal);       // strict < keeps first occurrence
      bestIdx = better ? idx  : bestIdx;
      bestVal = better ? cand : bestVal;
    }
  }

  // merge the two lane halves (codes 0-7 vs 8-15 of each tile)
  float oVal = __shfl_xor(bestVal, 16, 32);
  int   oIdx = __shfl_xor(bestIdx, 16, 32);
  if (oVal < bestVal || (oVal == bestVal && oIdx < bestIdx)) {
    bestVal = oVal; bestIdx = oIdx;
  }

  // gather winning code, write quantized (NCHW), accumulate SSE
  const float* qrow = w + bestIdx * D_DIM;
  float* obase = out + b * CH_IMG + hw;
  float sse = 0.f;
#pragma unroll
  for (int c = 0; c < 16; ++c) {
    int d0 = 4 * c + 2 * half;
    float q0 = qrow[d0], q1 = qrow[d0 + 1];
    obase[d0 * HW]       = q0;
    obase[(d0 + 1) * HW] = q1;
    float e0 = q0 - zb_x[c], e1 = q1 - zb_y[c];
    sse += e0 * e0 + e1 * e1;
  }
#pragma unroll
  for (int m = 16; m >= 1; m >>= 1) sse += __shfl_xor(sse, m, 32);
  if (lane == 0) {
    const float scale = 1.0f / (16384.0f * 64.0f);
    atomicAdd(out + OUT_ELEMS,     0.25f * scale * sse);  // commitment_loss
    atomicAdd(out + OUT_ELEMS + 1,         scale * sse);  // codebook_loss
  }
}

extern "C" void kernel_launch(void* const* d_in, const int* in_sizes, int n_in,
                              void* d_out, int out_size, void* d_ws, size_t ws_size,
                              hipStream_t stream) {
  const float* z = (const float*)d_in[0];   // (16,64,32,32) f32
  const float* w = (const float*)d_in[1];   // (8192,64) f32
  float* out   = (float*)d_out;             // 1048576 + 2 f32
  float* norms = (float*)d_ws;              // 8192 f32 scratch

  vq_zero_losses<<<1, 32, 0, stream>>>(out);
  vq_norms_kernel<<<K_CODES / 256, 256, 0, stream>>>(w, norms);
  // 16384 rows / 16 per wave = 1024 waves; 8 waves per 256-thread block -> 128 blocks
  vq_main_kernel<<<128, 256, 0, stream>>>(z, w, norms, out);
}